// GAT_73675868995821
// MI455X (gfx1250) — compile-verified
//
#include <hip/hip_runtime.h>

#define NNODES 50000
#define NEDGES 800000
#define FIN    256
#define NHEAD  8
#define FH     32      // per-head features, layer 1
#define NHID   256     // NHEAD * FH
#define NCLASS 47
#define NCPAD  48      // padded layer-2 width

typedef __attribute__((ext_vector_type(2))) float v2f;
typedef __attribute__((ext_vector_type(8))) float v8f;

// ---------------------------------------------------------------------------
// Layer-1 GEMM: h1[N,32] = x[N,256] @ W1[256,32], fp32 WMMA 16x16x4.
// One wave per 16-ROW strip; the wave owns BOTH 16-col tiles so the A panel
// (the 51 MB activation matrix) is read exactly once. 2 accumulators.
// A layout: lane l(0..15) holds row M=l, K = k0 + 2*half .. +1 (v2f).
// B layout: lane l holds col N=l, same K pair.
// D layout: VGPR r -> (M = r + 8*half, N = l).
// ---------------------------------------------------------------------------
__global__ void gemm_h1(const float* __restrict__ x, const float* __restrict__ W1,
                        float* __restrict__ h1) {
  const int wid = (blockIdx.x * blockDim.x + threadIdx.x) >> 5;
  if (wid >= NNODES / 16) return;
  const int rowBase = wid * 16;
  const int lane = threadIdx.x & 31;
  const int half = lane >> 4;
  const int l = lane & 15;
  const float* __restrict__ arow = x + (size_t)(rowBase + l) * FIN;
  v8f c0 = {}, c1 = {};
#pragma unroll 4
  for (int k0 = 0; k0 < FIN; k0 += 4) {
    const int k = k0 + 2 * half;
    const v2f a = *(const v2f*)(arow + k);            // 8B aligned: k even
    const float* __restrict__ w0 = W1 + k * FH;       // B row k
    const float* __restrict__ w1 = W1 + (k + 1) * FH; // B row k+1
    v2f b0, b1;
    b0.x = w0[l];      b0.y = w1[l];
    b1.x = w0[16 + l]; b1.y = w1[16 + l];
    c0 = __builtin_amdgcn_wmma_f32_16x16x4_f32(false, a, false, b0,
                                               (short)0, c0, false, false);
    c1 = __builtin_amdgcn_wmma_f32_16x16x4_f32(false, a, false, b1,
                                               (short)0, c1, false, false);
  }
#pragma unroll
  for (int r = 0; r < 8; ++r) {
    const int row = rowBase + r + 8 * half;
    h1[(size_t)row * FH + l]      = c0[r];
    h1[(size_t)row * FH + 16 + l] = c1[r];
  }
}

// ---------------------------------------------------------------------------
// Layer-2 GEMM: h2[N,48] = z[N,256] @ W2[256,47] (col 47 zero-padded).
// One wave per 16-row strip owning all THREE col tiles: A read once.
// ---------------------------------------------------------------------------
__global__ void gemm_h2(const float* __restrict__ z, const float* __restrict__ W2,
                        float* __restrict__ h2) {
  const int wid = (blockIdx.x * blockDim.x + threadIdx.x) >> 5;
  if (wid >= NNODES / 16) return;
  const int rowBase = wid * 16;
  const int lane = threadIdx.x & 31;
  const int half = lane >> 4;
  const int l = lane & 15;
  const float* __restrict__ arow = z + (size_t)(rowBase + l) * NHID;
  v8f c0 = {}, c1 = {}, c2 = {};
#pragma unroll 4
  for (int k0 = 0; k0 < NHID; k0 += 4) {
    const int k = k0 + 2 * half;
    const v2f a = *(const v2f*)(arow + k);
    const float* __restrict__ w0 = W2 + k * NCLASS;
    const float* __restrict__ w1 = W2 + (k + 1) * NCLASS;
    v2f b0, b1, b2;
    b0.x = w0[l];      b0.y = w1[l];
    b1.x = w0[16 + l]; b1.y = w1[16 + l];
    b2.x = (32 + l < NCLASS) ? w0[32 + l] : 0.0f;
    b2.y = (32 + l < NCLASS) ? w1[32 + l] : 0.0f;
    c0 = __builtin_amdgcn_wmma_f32_16x16x4_f32(false, a, false, b0,
                                               (short)0, c0, false, false);
    c1 = __builtin_amdgcn_wmma_f32_16x16x4_f32(false, a, false, b1,
                                               (short)0, c1, false, false);
    c2 = __builtin_amdgcn_wmma_f32_16x16x4_f32(false, a, false, b2,
                                               (short)0, c2, false, false);
  }
#pragma unroll
  for (int r = 0; r < 8; ++r) {
    const int row = rowBase + r + 8 * half;
    float* __restrict__ orow = h2 + (size_t)row * NCPAD;
    orow[l]      = c0[r];
    orow[16 + l] = c1[r];
    orow[32 + l] = c2[r];   // col 47 stores 0 (B padded)
  }
}

// ---------------------------------------------------------------------------
// Attention scores, layer 1: el/er[N,8] = h1 @ Wl1/Wr1 (32x8 each)
// ---------------------------------------------------------------------------
__global__ void att1(const float* __restrict__ h1, const float* __restrict__ Wl,
                     const float* __restrict__ Wr, float* __restrict__ el,
                     float* __restrict__ er) {
  const int idx = blockIdx.x * blockDim.x + threadIdx.x;
  if (idx >= NNODES * NHEAD) return;
  const int n = idx >> 3, h = idx & 7;
  const float* __restrict__ hp = h1 + (size_t)n * FH;
  float sl = 0.f, sr = 0.f;
#pragma unroll
  for (int f = 0; f < FH; ++f) {
    const float hv = hp[f];
    sl += hv * Wl[f * NHEAD + h];
    sr += hv * Wr[f * NHEAD + h];
  }
  el[idx] = sl;
  er[idx] = sr;
}

// ---------------------------------------------------------------------------
// Attention scores, layer 2 (single head): el2/er2[N]
// ---------------------------------------------------------------------------
__global__ void att2(const float* __restrict__ h2, const float* __restrict__ Wl2,
                     const float* __restrict__ Wr2, float* __restrict__ el2,
                     float* __restrict__ er2) {
  const int n = blockIdx.x * blockDim.x + threadIdx.x;
  if (n >= NNODES) return;
  const float* __restrict__ hp = h2 + (size_t)n * NCPAD;
  float sl = 0.f, sr = 0.f;
#pragma unroll
  for (int c = 0; c < NCLASS; ++c) {
    const float v = hp[c];
    sl += v * Wl2[c];
    sr += v * Wr2[c];
  }
  el2[n] = sl;
  er2[n] = sr;
}

// ---------------------------------------------------------------------------
// Edge phase, layer 1. One wave per edge.
// lanes 0..7: per-head attention weight + denom atomic; then all lanes
// scatter-add 8 heads x 32 features using shfl broadcast of the head weight.
// agg layout: [N, head, feat] flat = [N, 256]. All atomics resolve in L2.
// ---------------------------------------------------------------------------
__global__ void edge1(const int* __restrict__ src, const int* __restrict__ dst,
                      const float* __restrict__ el, const float* __restrict__ er,
                      const float* __restrict__ h1, float* __restrict__ denom,
                      float* __restrict__ agg) {
  const int wid = (blockIdx.x * blockDim.x + threadIdx.x) >> 5;
  if (wid >= NEDGES) return;
  const int lane = threadIdx.x & 31;
  const int s = src[wid];
  const int d = dst[wid];
  float a = 0.f;
  if (lane < NHEAD) {
    float z = el[s * NHEAD + lane] + er[d * NHEAD + lane];
    z = (z > 0.f) ? z : 0.2f * z;          // leaky_relu(0.2)
    a = __expf(z);
    atomicAdd(&denom[s * NHEAD + lane], a);
  }
  const float hv = h1[(size_t)d * FH + lane];   // feat = lane
  float* __restrict__ ap = agg + (size_t)s * NHID;
#pragma unroll
  for (int h = 0; h < NHEAD; ++h) {
    const float ah = __shfl(a, h, 32);
    atomicAdd(&ap[h * FH + lane], ah * hv);
  }
}

// ---------------------------------------------------------------------------
// Edge phase, layer 2 (single head, 47 features). One wave per edge.
// ---------------------------------------------------------------------------
__global__ void edge2(const int* __restrict__ src, const int* __restrict__ dst,
                      const float* __restrict__ el2, const float* __restrict__ er2,
                      const float* __restrict__ h2, float* __restrict__ denom2,
                      float* __restrict__ agg2) {
  const int wid = (blockIdx.x * blockDim.x + threadIdx.x) >> 5;
  if (wid >= NEDGES) return;
  const int lane = threadIdx.x & 31;
  const int s = src[wid];
  const int d = dst[wid];
  float z = el2[s] + er2[d];
  z = (z > 0.f) ? z : 0.2f * z;
  const float a = __expf(z);
  if (lane == 0) atomicAdd(&denom2[s], a);
  const float* __restrict__ hp = h2 + (size_t)d * NCPAD;
  float* __restrict__ ap = agg2 + (size_t)s * NCPAD;
  for (int f = lane; f < NCLASS; f += 32) atomicAdd(&ap[f], a * hp[f]);
}

// ---------------------------------------------------------------------------
// Finalize layer 1 (in place): agg = elu(agg/max(denom,1e-12) + b1[f])
// ---------------------------------------------------------------------------
__global__ void finalize1(float* __restrict__ agg, const float* __restrict__ denom,
                          const float* __restrict__ b1) {
  const size_t idx = (size_t)blockIdx.x * blockDim.x + threadIdx.x;
  if (idx >= (size_t)NNODES * NHID) return;
  const int n = (int)(idx >> 8);
  const int j = (int)(idx & 255);
  const int h = j >> 5;
  const int f = j & 31;
  const float v = agg[idx] / fmaxf(denom[n * NHEAD + h], 1e-12f) + b1[f];
  agg[idx] = (v > 0.f) ? v : expm1f(v);   // elu
}

// ---------------------------------------------------------------------------
// Finalize layer 2 + log_softmax. One wave per node; shfl_xor reductions.
// ---------------------------------------------------------------------------
__global__ void finalize2(const float* __restrict__ agg2,
                          const float* __restrict__ denom2,
                          const float* __restrict__ b2, float* __restrict__ out) {
  const int n = (blockIdx.x * blockDim.x + threadIdx.x) >> 5;
  if (n >= NNODES) return;
  const int lane = threadIdx.x & 31;
  const float den = fmaxf(denom2[n], 1e-12f);
  const float* __restrict__ ap = agg2 + (size_t)n * NCPAD;
  const float v0 = ap[lane] / den + b2[lane];
  const bool hasHi = lane < (NCLASS - 32);
  const float v1 = hasHi ? (ap[lane + 32] / den + b2[lane + 32]) : -3.4e38f;
  float m = fmaxf(v0, v1);
#pragma unroll
  for (int off = 16; off > 0; off >>= 1) m = fmaxf(m, __shfl_xor(m, off, 32));
  float ssum = __expf(v0 - m) + (hasHi ? __expf(v1 - m) : 0.f);
#pragma unroll
  for (int off = 16; off > 0; off >>= 1) ssum += __shfl_xor(ssum, off, 32);
  const float lse = m + __logf(ssum);
  float* __restrict__ op = out + (size_t)n * NCLASS;
  op[lane] = v0 - lse;
  if (hasHi) op[lane + 32] = v1 - lse;
}

// ---------------------------------------------------------------------------
extern "C" void kernel_launch(void* const* d_in, const int* in_sizes, int n_in,
                              void* d_out, int out_size, void* d_ws, size_t ws_size,
                              hipStream_t stream) {
  (void)in_sizes; (void)n_in; (void)out_size; (void)ws_size;
  const float* x    = (const float*)d_in[0];
  const int*   esrc = (const int*)d_in[1];
  const int*   edst = (const int*)d_in[2];
  const float* W1   = (const float*)d_in[3];
  const float* Wl1  = (const float*)d_in[4];
  const float* Wr1  = (const float*)d_in[5];
  const float* b1   = (const float*)d_in[6];
  const float* W2   = (const float*)d_in[7];
  const float* Wl2  = (const float*)d_in[8];
  const float* Wr2  = (const float*)d_in[9];
  const float* b2   = (const float*)d_in[10];
  float* out = (float*)d_out;

  // Workspace layout: zero-initialized atomic targets first (single memset).
  float* ws = (float*)d_ws;
  size_t off = 0;
  float* agg1   = ws + off; off += (size_t)NNODES * NHID;   // 12.8M
  float* denom1 = ws + off; off += (size_t)NNODES * NHEAD;  // 0.4M
  float* agg2   = ws + off; off += (size_t)NNODES * NCPAD;  // 2.4M
  float* denom2 = ws + off; off += (size_t)NNODES;          // 50K
  const size_t zeroFloats = off;
  float* h1v = ws + off; off += (size_t)NNODES * FH;        // 1.6M
  float* el1 = ws + off; off += (size_t)NNODES * NHEAD;
  float* er1 = ws + off; off += (size_t)NNODES * NHEAD;
  float* h2v = ws + off; off += (size_t)NNODES * NCPAD;
  float* el2 = ws + off; off += (size_t)NNODES;
  float* er2 = ws + off; off += (size_t)NNODES;

  hipMemsetAsync(d_ws, 0, zeroFloats * sizeof(float), stream);

  // ---- Layer 1 ----
  {
    const int strips = NNODES / 16;   // 3125 waves
    gemm_h1<<<(strips * 32 + 255) / 256, 256, 0, stream>>>(x, W1, h1v);
  }
  att1<<<(NNODES * NHEAD + 255) / 256, 256, 0, stream>>>(h1v, Wl1, Wr1, el1, er1);
  edge1<<<(NEDGES * 32 + 255) / 256, 256, 0, stream>>>(esrc, edst, el1, er1, h1v,
                                                       denom1, agg1);
  {
    const size_t tot = (size_t)NNODES * NHID;
    finalize1<<<(unsigned)((tot + 255) / 256), 256, 0, stream>>>(agg1, denom1, b1);
  }

  // ---- Layer 2 (input = agg1 in place) ----
  {
    const int strips = NNODES / 16;   // 3125 waves
    gemm_h2<<<(strips * 32 + 255) / 256, 256, 0, stream>>>(agg1, W2, h2v);
  }
  att2<<<(NNODES + 255) / 256, 256, 0, stream>>>(h2v, Wl2, Wr2, el2, er2);
  edge2<<<(NEDGES * 32 + 255) / 256, 256, 0, stream>>>(esrc, edst, el2, er2, h2v,
                                                       denom2, agg2);
  finalize2<<<(NNODES * 32 + 255) / 256, 256, 0, stream>>>(agg2, denom2, b2, out);
}